// R3KVHadamard_36361193128489
// MI455X (gfx1250) — compile-verified
//
#include <hip/hip_runtime.h>

typedef float v2f __attribute__((ext_vector_type(2)));
typedef float v4f __attribute__((ext_vector_type(4)));
typedef float v8f __attribute__((ext_vector_type(8)));

#define ROWSTRIDE 132   // 128 + 4 dword pad: r*132 mod 64 = 4r -> spreads banks

__device__ __forceinline__ float hsgn(int a, int b) {
    return (__popc(a & b) & 1) ? -1.0f : 1.0f;
}

// One 16x128 tile per wave. Y = (H16 (x) H8) * x per 128-block, scale folded
// into the stage-2 constant A operand. Single LDS buffer: input -> T -> Y
// all in place (stage-2 column j of T is dead once column j of Y is formed).
__global__ __launch_bounds__(64) void
R3KVHadamard_36361193128489_kernel(const float* __restrict__ src,
                                   float* __restrict__ dst) {
    __shared__ float ldsT[2][16 * ROWSTRIDE];   // input tile, then T, then Y

    const int wave = threadIdx.x >> 5;
    const int lane = threadIdx.x & 31;
    const int n    = lane & 15;   // M index (A) / N index (B,D)
    const int hi   = lane >> 4;   // selects K sub-pair {0,1} vs {2,3}

    float* bufT = ldsT[wave];

    const size_t tile = (size_t)blockIdx.x * 2 + wave;
    const size_t base = tile * (16u * 128u);

    // ---- Phase 1: coalesced streaming global -> LDS (padded rows) ---------
    #pragma unroll
    for (int i = 0; i < 4; ++i) {
        int chunk = lane + i * 32;        // 0..127 float4 chunks of the tile
        int row = chunk >> 3;
        int c4  = chunk & 7;
        v4f d = __builtin_nontemporal_load((const v4f*)(src + base + (size_t)chunk * 4));
        *(v4f*)(bufT + row * ROWSTRIDE + c4 * 4) = d;
    }
    __syncthreads();

    // ---- Constant A operands (generated in-register) ----------------------
    // A-layout (32-bit 16x4): lanes 0-15 hold K={k0,k0+1}, lanes 16-31 K={k0+2,k0+3}
    const float scale = 0.08838834764831845f;   // 1/sqrt(128)
    v2f a8[4], a16[4];
    #pragma unroll
    for (int kc = 0; kc < 4; ++kc) {
        int k = kc * 4 + 2 * hi;
        // stage 1: diag(H8, H8)
        v2f t;
        t.x = ((n ^ k)       & 8) ? 0.0f : hsgn(n & 7, k & 7);
        t.y = ((n ^ (k + 1)) & 8) ? 0.0f : hsgn(n & 7, (k + 1) & 7);
        a8[kc] = t;
        // stage 2: H16 * (1/sqrt(128))
        v2f u;
        u.x = hsgn(n, k) * scale;
        u.y = hsgn(n, k + 1) * scale;
        a16[kc] = u;
    }

    // ---- Stage 1: T[r, 2pp+s, q] = sum_q' H8[q,q'] X[r, (2pp+s)*8+q'] -----
    // B_pp[k,n] = X[n, pp*16 + k]  (contiguous in LDS -> b64 loads)
    v8f T[8];
    #pragma unroll
    for (int pp = 0; pp < 8; ++pp) {
        v8f acc = {};
        #pragma unroll
        for (int kc = 0; kc < 4; ++kc) {
            int k = kc * 4 + 2 * hi;
            v2f b = *(const v2f*)(bufT + n * ROWSTRIDE + pp * 16 + k);
            acc = __builtin_amdgcn_wmma_f32_16x16x4_f32(
                      false, a8[kc], false, b, (short)0, acc, false, false);
        }
        T[pp] = acc;
    }
    __syncthreads();

    // D layout: VGPR v, lanes0-15 -> T[r=n, p=2pp,   q=v]
    //                   lanes16-31-> T[r=n, p=2pp+1, q=v]
    #pragma unroll
    for (int pp = 0; pp < 8; ++pp) {
        float* p0 = bufT + n * ROWSTRIDE + (2 * pp + hi) * 8;
        *(v4f*)(p0)     = *((const v4f*)&T[pp]);
        *(v4f*)(p0 + 4) = *((const v4f*)&T[pp] + 1);
    }
    __syncthreads();

    // ---- Stage 2 (in place): Y[r, p*8+j] = s * sum_p' H16[p,p'] T[r,p',j] -
    // B2_j[k=p', n=r] = T[r, p', j]. Column j of T is fully consumed by the
    // loads of iteration j before the stores of iteration j (per-wave LDS
    // ops execute in program order), so Y overwrites T column-by-column.
    #pragma unroll
    for (int j = 0; j < 8; ++j) {
        v8f acc = {};
        #pragma unroll
        for (int kc = 0; kc < 4; ++kc) {
            int k = kc * 4 + 2 * hi;
            v2f b;
            b.x = bufT[n * ROWSTRIDE + k * 8 + j];
            b.y = bufT[n * ROWSTRIDE + (k + 1) * 8 + j];
            acc = __builtin_amdgcn_wmma_f32_16x16x4_f32(
                      false, a16[kc], false, b, (short)0, acc, false, false);
        }
        // D: VGPR v, lanes0-15 -> Y[r=n, c=v*8+j]; lanes16-31 -> c=(v+8)*8+j
        #pragma unroll
        for (int v = 0; v < 8; ++v) {
            bufT[n * ROWSTRIDE + (v + 8 * hi) * 8 + j] = acc[v];
        }
    }
    __syncthreads();

    // ---- Phase 4: coalesced LDS -> streaming global store -----------------
    #pragma unroll
    for (int i = 0; i < 4; ++i) {
        int chunk = lane + i * 32;
        int row = chunk >> 3;
        int c4  = chunk & 7;
        v4f d = *(const v4f*)(bufT + row * ROWSTRIDE + c4 * 4);
        __builtin_nontemporal_store(d, (v4f*)(dst + base + (size_t)chunk * 4));
    }
}

extern "C" void kernel_launch(void* const* d_in, const int* in_sizes, int n_in,
                              void* d_out, int out_size, void* d_ws, size_t ws_size,
                              hipStream_t stream) {
    (void)d_ws; (void)ws_size; (void)out_size;
    size_t off = 0;
    for (int i = 0; i < n_in && i < 3; ++i) {
        const float* src = (const float*)d_in[i];
        float* dst = (float*)d_out + off;
        size_t nTiles = (size_t)in_sizes[i] / (16 * 128);  // 16-row tiles
        int blocks = (int)(nTiles / 2);                    // 2 waves (tiles) / block
        R3KVHadamard_36361193128489_kernel<<<blocks, 64, 0, stream>>>(src, dst);
        off += (size_t)in_sizes[i];
    }
}